// PolyGIoULoss_31061203485091
// MI455X (gfx1250) — compile-verified
//
#include <hip/hip_runtime.h>
#include <math.h>

#define GEPS 1e-6f

typedef float v2f __attribute__((ext_vector_type(2)));
typedef float v8f __attribute__((ext_vector_type(8)));

__device__ __forceinline__ void box2corners(float x, float y, float w, float h, float a,
                                            float cx[4], float cy[4]) {
    float s, c;
    sincosf(a, &s, &c);
    const float dx[4] = {0.5f, -0.5f, -0.5f, 0.5f};
    const float dy[4] = {0.5f, 0.5f, -0.5f, -0.5f};
#pragma unroll
    for (int k = 0; k < 4; ++k) {
        float ox = dx[k] * w, oy = dy[k] * h;
        cx[k] = ox * c - oy * s + x;
        cy[k] = ox * s + oy * c + y;
    }
}

// mask for each corner (qx,qy) lying inside the box whose corners are (bx,by)
__device__ __forceinline__ void corners_in_box(const float qx[4], const float qy[4],
                                               const float bx[4], const float by[4],
                                               bool m[4]) {
    const float tol = 1e-6f;
    float ax = bx[0], ay = by[0];
    float abx = bx[1] - ax, aby = by[1] - ay;
    float adx = bx[3] - ax, ady = by[3] - ay;
    float dab = abx * abx + aby * aby;
    float dad = adx * adx + ady * ady;
#pragma unroll
    for (int k = 0; k < 4; ++k) {
        float amx = qx[k] - ax, amy = qy[k] - ay;
        float p_ab = (amx * abx + amy * aby) / dab;
        float p_ad = (amx * adx + amy * ady) / dad;
        m[k] = (p_ab > -tol) && (p_ab < 1.f + tol) && (p_ad > -tol) && (p_ad < 1.f + tol);
    }
}

// Sort-by-angle star polygon area. Mutates px/py. Matches reference semantics:
// masked points get ang=1e9 (sort to the end, stable), tail replaced by first
// sorted point, shoelace, abs, *0.5.
template <int NP>
__device__ __forceinline__ float sorted_poly_area(float (&px)[NP], float (&py)[NP],
                                                  const bool (&m)[NP]) {
    float cntf = 0.f, sx = 0.f, sy = 0.f;
#pragma unroll
    for (int k = 0; k < NP; ++k) {
        float mk = m[k] ? 1.f : 0.f;
        cntf += mk;
        sx += px[k] * mk;
        sy += py[k] * mk;
    }
    float num = fmaxf(cntf, 1.f);
    float mx = sx / num, my = sy / num;

    float ang[NP];
#pragma unroll
    for (int k = 0; k < NP; ++k)
        ang[k] = m[k] ? atan2f(py[k] - my, px[k] - mx) : 1e9f;

    // odd-even transposition sort: stable (adjacent swaps only on strict <)
#pragma unroll
    for (int pass = 0; pass < NP; ++pass) {
#pragma unroll
        for (int j = (pass & 1); j + 1 < NP; j += 2) {
            bool sw = ang[j + 1] < ang[j];
            float t;
            t = ang[j]; ang[j] = sw ? ang[j + 1] : ang[j]; ang[j + 1] = sw ? t : ang[j + 1];
            t = px[j];  px[j]  = sw ? px[j + 1]  : px[j];  px[j + 1]  = sw ? t : px[j + 1];
            t = py[j];  py[j]  = sw ? py[j + 1]  : py[j];  py[j + 1]  = sw ? t : py[j + 1];
        }
    }

    int cnt = (int)cntf;
    float x0 = px[0], y0 = py[0];
#pragma unroll
    for (int k = 0; k < NP; ++k) {
        bool v = k < cnt;
        px[k] = v ? px[k] : x0;
        py[k] = v ? py[k] : y0;
    }

    float s = 0.f;
#pragma unroll
    for (int k = 0; k < NP; ++k) {
        int kn = (k + 1 == NP) ? 0 : k + 1;
        s += px[k] * py[kn] - py[k] * px[kn];
    }
    return 0.5f * fabsf(s);
}

__global__ void __launch_bounds__(256)
poly_giou_partial_kernel(const float* __restrict__ pred, const float* __restrict__ tgt,
                         const float* __restrict__ wgt, float* __restrict__ partial, int N) {
    int i = blockIdx.x * 256 + threadIdx.x;
    float val = 0.f;
    if (i < N) {
        const float* p = pred + 5 * (size_t)i;
        const float* q = tgt + 5 * (size_t)i;
        float p_x = p[0], p_y = p[1], p_w = p[2], p_h = p[3], p_a = p[4];
        float q_x = q[0], q_y = q[1], q_w = q[2], q_h = q[3], q_a = q[4];

        float c1x[4], c1y[4], c2x[4], c2y[4];
        box2corners(p_x, p_y, p_w, p_h, p_a, c1x, c1y);
        box2corners(q_x, q_y, q_w, q_h, q_a, c2x, c2y);

        float ptx[24], pty[24];
        bool msk[24];
        bool m1[4], m2[4];
        corners_in_box(c1x, c1y, c2x, c2y, m1);
        corners_in_box(c2x, c2y, c1x, c1y, m2);
#pragma unroll
        for (int k = 0; k < 4; ++k) {
            ptx[k] = c1x[k];     pty[k] = c1y[k];     msk[k] = m1[k];
            ptx[4 + k] = c2x[k]; pty[4 + k] = c2y[k]; msk[4 + k] = m2[k];
        }

        // 4x4 segment intersections (edges roll(-1) like the reference)
#pragma unroll
        for (int j = 0; j < 4; ++j) {
            float x1 = c1x[j], y1 = c1y[j];
            float x2 = c1x[(j + 1) & 3], y2 = c1y[(j + 1) & 3];
#pragma unroll
            for (int k = 0; k < 4; ++k) {
                float x3 = c2x[k], y3 = c2y[k];
                float x4 = c2x[(k + 1) & 3], y4 = c2y[(k + 1) & 3];
                float den = (x2 - x1) * (y4 - y3) - (y2 - y1) * (x4 - x3);
                float dsafe = (den == 0.f) ? 1.f : den;
                float t = ((x3 - x1) * (y4 - y3) - (y3 - y1) * (x4 - x3)) / dsafe;
                float u = ((x3 - x1) * (y2 - y1) - (y3 - y1) * (x2 - x1)) / dsafe;
                bool mk = (den != 0.f) && (t > 0.f) && (t < 1.f) && (u > 0.f) && (u < 1.f);
                int idx = 8 + j * 4 + k;
                ptx[idx] = mk ? (x1 + t * (x2 - x1)) : 0.f;
                pty[idx] = mk ? (y1 + t * (y2 - y1)) : 0.f;
                msk[idx] = mk;
            }
        }

        float overlap = sorted_poly_area<24>(ptx, pty, msk);

        float ex[8], ey[8];
        bool em[8];
#pragma unroll
        for (int k = 0; k < 4; ++k) {
            ex[k] = c1x[k];     ey[k] = c1y[k];     em[k] = true;
            ex[4 + k] = c2x[k]; ey[4 + k] = c2y[k]; em[4 + k] = true;
        }
        float enclose = sorted_poly_area<8>(ex, ey, em);

        float a1 = p_w * p_h;
        float a2 = q_w * q_h;
        float uni = a1 + a2 - overlap + GEPS;
        float iou = fmaxf(overlap / uni, GEPS);
        float giou = iou - (enclose - uni) / enclose;
        float loss = 1.f - giou;
        val = loss * wgt[i];
    }

    __shared__ float red[256];
    red[threadIdx.x] = val;
    __syncthreads();
#pragma unroll
    for (int s = 128; s > 0; s >>= 1) {
        if (threadIdx.x < s) red[threadIdx.x] += red[threadIdx.x + s];
        __syncthreads();
    }
    if (threadIdx.x == 0) partial[blockIdx.x] = red[0];
}

// Final reduce: 256 strided accumulators -> LDS -> wave0 sums all 256 values
// with chained V_WMMA_F32_16X16X4_F32 (A = 64 partials, B = ones, C = acc).
// D[m][n] = rowsum(m) for all n; lane0 holds D[0..7][0], lane16 holds D[8..15][0].
__global__ void __launch_bounds__(256)
poly_giou_reduce_kernel(const float* __restrict__ partial, int nb,
                        float* __restrict__ out, float inv_n) {
    __shared__ float buf[256];
    float s = 0.f;
    for (int k = threadIdx.x; k < nb; k += 256) s += partial[k];
    buf[threadIdx.x] = s;
    __syncthreads();

    if (threadIdx.x < 32) {
        int lane = threadIdx.x;
        v8f acc = {};
        v2f ones;
        ones.x = 1.0f;
        ones.y = 1.0f;
#pragma unroll
        for (int t = 0; t < 4; ++t) {
            v2f a;
            a.x = buf[t * 64 + lane];
            a.y = buf[t * 64 + 32 + lane];
            acc = __builtin_amdgcn_wmma_f32_16x16x4_f32(
                /*neg_a=*/false, a, /*neg_b=*/false, ones,
                /*c_mod=*/(short)0, acc, /*reuse_a=*/false, /*reuse_b=*/false);
        }
        float s8 = acc[0] + acc[1] + acc[2] + acc[3] + acc[4] + acc[5] + acc[6] + acc[7];
        float tot = __shfl(s8, 0, 32) + __shfl(s8, 16, 32);
        if (lane == 0) out[0] = tot * inv_n;
    }
}

extern "C" void kernel_launch(void* const* d_in, const int* in_sizes, int n_in,
                              void* d_out, int out_size, void* d_ws, size_t ws_size,
                              hipStream_t stream) {
    const float* pred = (const float*)d_in[0];
    const float* tgt = (const float*)d_in[1];
    const float* wgt = (const float*)d_in[2];
    float* out = (float*)d_out;
    float* partial = (float*)d_ws;
    int N = in_sizes[2];
    int nb = (N + 255) / 256;
    poly_giou_partial_kernel<<<nb, 256, 0, stream>>>(pred, tgt, wgt, partial, N);
    poly_giou_reduce_kernel<<<1, 256, 0, stream>>>(partial, nb, out, 1.0f / (float)N);
}